// LogSig_rolling_1614907703722
// MI455X (gfx1250) — compile-verified
//
#include <hip/hip_runtime.h>

typedef float v2f __attribute__((ext_vector_type(2)));
typedef float v8f __attribute__((ext_vector_type(8)));

#define BB 256
#define NT 4096
#define DD 12
#define KW 64
#define STR 16
#define WW 253        // (NT - KW)/STR + 1
#define OUTD 78       // 12 + 66
#define WPB 8         // waves per block

__global__ __launch_bounds__(256) void logsig_wmma_kernel(
    const float* __restrict__ inp, float* __restrict__ out) {
  // per-wave: 64 rows x 16 floats (padded window) + 16x16 M matrix
  __shared__ float sWin[WPB * KW * 16];   // 8 * 1024 floats = 32 KB
  __shared__ float sM[WPB * 256];         // 8 *  256 floats =  8 KB

  const int tid  = threadIdx.x;
  const int lane = tid & 31;
  const int wid  = tid >> 5;
  const int gw   = blockIdx.x * WPB + wid;     // global window id, exact fit
  const int b    = gw / WW;
  const int w    = gw - b * WW;

  const float* gsrc = inp + ((size_t)b * NT + (size_t)w * STR) * DD;
  float* sw = &sWin[wid * KW * 16];
  float* sm = &sM[wid * 256];

  // Load 64 rows x 12 floats (3 aligned float4 per row) into 16-float rows.
  for (int t = lane; t < KW * 3; t += 32) {
    const int row = t / 3, ch = t - row * 3;
    const float4 v = *reinterpret_cast<const float4*>(gsrc + row * DD + ch * 4);
    *reinterpret_cast<float4*>(sw + row * 16 + ch * 4) = v;
  }
  // Zero pad columns 12..15 so M/N rows 12-15 are exactly zero.
  for (int r = lane; r < KW; r += 32) {
    *reinterpret_cast<float4*>(sw + r * 16 + 12) = make_float4(0.f, 0.f, 0.f, 0.f);
  }
  __syncthreads();

  // Operand indexing per CDNA5 32-bit A(16x4)/B(4x16) layouts:
  //   lane = (m or n) + 16*(k/2), vgpr = k%2
  const int m  = lane & 15;
  const int kh = lane >> 4;                 // K-half selector (0 or 1)
  const float w0 = sw[m];                   // win[0][m]

  v8f acc = {0.f, 0.f, 0.f, 0.f, 0.f, 0.f, 0.f, 0.f};
#pragma unroll
  for (int k0 = 0; k0 < KW; k0 += 4) {
    const int kb = k0 + 2 * kh;             // first k this lane provides
    const float wa = sw[kb * 16 + m];       // win[kb]
    const float wb = sw[(kb + 1) * 16 + m]; // win[kb+1]
    const int  k2 = (kb + 2 > KW - 1) ? (KW - 1) : (kb + 2);
    const float wc = sw[k2 * 16 + m];       // win[kb+2] (clamped)
    v2f a, bmat;
    a[0]    = wa - w0;                      // cum[kb]   = win[kb]   - win[0]
    a[1]    = wb - w0;                      // cum[kb+1] = win[kb+1] - win[0]
    bmat[0] = wb - wa;                      // delta[kb]
    bmat[1] = (kb == KW - 2) ? 0.f : (wc - wb);  // delta[63] padded to 0
    // D = A(16x4) * B(4x16) + C : v_wmma_f32_16x16x4_f32
    acc = __builtin_amdgcn_wmma_f32_16x16x4_f32(
        /*neg_a=*/false, a, /*neg_b=*/false, bmat,
        /*c_mod=*/(short)0, acc, /*reuse_a=*/false, /*reuse_b=*/false);
  }

  // Spill M to LDS: C/D layout row = v + 8*(lane/16), col = lane%16.
#pragma unroll
  for (int v = 0; v < 8; ++v) {
    sm[(v + 8 * kh) * 16 + m] = acc[v];
  }
  __syncthreads();

  float* gout = out + ((size_t)b * WW + w) * OUTD;
  // lvl1 = win[63] - win[0]
  if (lane < DD) {
    gout[lane] = sw[(KW - 1) * 16 + lane] - sw[lane];
  }
  // A[i,j] = 0.5*(M[i,j] - M[j,i]) for the 66 strict upper-triangular pairs.
  for (int t = lane; t < 66; t += 32) {
    int i = 0, off = 0, len = DD - 1;
    while (t >= off + len) { off += len; --len; ++i; }
    const int j = i + 1 + (t - off);
    gout[DD + t] = 0.5f * (sm[i * 16 + j] - sm[j * 16 + i]);
  }
}

extern "C" void kernel_launch(void* const* d_in, const int* in_sizes, int n_in,
                              void* d_out, int out_size, void* d_ws, size_t ws_size,
                              hipStream_t stream) {
  (void)in_sizes; (void)n_in; (void)d_ws; (void)ws_size; (void)out_size;
  const float* inp = (const float*)d_in[0];
  float* out = (float*)d_out;
  // 256*253 = 64768 windows, one wave each, 8 waves per 256-thread block.
  dim3 grid((BB * WW) / WPB);   // 8096 blocks, exact — every wave full
  dim3 block(256);
  logsig_wmma_kernel<<<grid, block, 0, stream>>>(inp, out);
}